// Layer_87935160418772
// MI455X (gfx1250) — compile-verified
//
#include <hip/hip_runtime.h>
#include <math.h>

#define IN_DIM 128
#define DIM_K  32
#define FAC_K  4
#define NSLOPE 0.2f

typedef __attribute__((ext_vector_type(2))) float v2f;
typedef __attribute__((ext_vector_type(8))) float v8f;

// ---------------------------------------------------------------------------
// sum across all 32 lanes of a wave (wave32 on gfx1250)
__device__ __forceinline__ float waveSum32(float v) {
#pragma unroll
    for (int off = 16; off > 0; off >>= 1)
        v += __shfl_xor(v, off, 32);
    return v;
}

// ---------------------------------------------------------------------------
// fac[n*128 + f*32 + c] = leaky_relu( sum_d allEmb[n,d] * (W[f,d,c] + b[f,c]) )
// One wave = one 16x16 output tile (f, colhalf fixed per wave).
// Uses V_WMMA_F32_16X16X4_F32: A 16x4 f32 (2 VGPR), B 4x16 f32 (2 VGPR), C 8 VGPR.
__global__ void gemm_fac_kernel(const float* __restrict__ allEmb,
                                const float* __restrict__ W,
                                const float* __restrict__ B,
                                float* __restrict__ fac,
                                int nRowTiles)
{
    const int lane   = threadIdx.x & 31;
    const int wave   = threadIdx.x >> 5;     // 0..7 -> {factor, colhalf}
    const int rowTile = blockIdx.x;
    if (rowTile >= nRowTiles) return;

    const int f      = wave >> 1;
    const int ch     = wave & 1;
    const int laneM  = lane & 15;
    const int laneHi = lane >> 4;            // 0: K/M low half, 1: high half
    const int row0   = rowTile * 16;
    const int col    = ch * 16 + laneM;      // column within this factor (0..31)

    const float bias = B[f * DIM_K + col];

    // A fragment source: allEmb[row0+laneM, k + laneHi*2 .. +1]
    const float* aRow = allEmb + (size_t)(row0 + laneM) * IN_DIM + laneHi * 2;
    // B fragment source: W[f, k + laneHi*2 .. +1, col]
    const float* wBase = W + (size_t)f * IN_DIM * DIM_K + (size_t)(laneHi * 2) * DIM_K + col;

    v8f c = {};
#pragma unroll 4
    for (int k = 0; k < IN_DIM; k += 4) {
        v2f a = *(const v2f*)(aRow + k);                 // 8B aligned
        v2f bb;
        bb.x = wBase[(size_t)k * DIM_K]       + bias;
        bb.y = wBase[(size_t)(k + 1) * DIM_K] + bias;
        c = __builtin_amdgcn_wmma_f32_16x16x4_f32(
                /*neg_a=*/false, a, /*neg_b=*/false, bb,
                /*c_mod=*/(short)0, c, /*reuse_a=*/false, /*reuse_b=*/false);
    }

    // D layout: lanes 0-15 -> M=r, lanes 16-31 -> M=r+8 ; N = laneM
#pragma unroll
    for (int r = 0; r < 8; ++r) {
        const int m = laneHi * 8 + r;
        float v = c[r];
        v = (v >= 0.f) ? v : NSLOPE * v;                 // leaky_relu(0.2)
        fac[(size_t)(row0 + m) * (FAC_K * DIM_K) + f * DIM_K + col] = v;
    }
}

// ---------------------------------------------------------------------------
// L2-normalize each (n,f) 32-vector of fac in place, and copy to cur.
// Each 32-lane group covers exactly one (n,f) chunk (layout is 32-aligned).
__global__ void norm_init_kernel(float* __restrict__ fac,
                                 float* __restrict__ cur)
{
    const size_t idx = (size_t)blockIdx.x * blockDim.x + threadIdx.x;
    float x = fac[idx];
    float s = waveSum32(x * x);
    float y = x / fmaxf(sqrtf(s), 1e-12f);
    fac[idx] = y;
    cur[idx] = y;
}

// ---------------------------------------------------------------------------
// cur[idx] = l2norm_over_32( fac[idx] + agg[idx] )
__global__ void combine_norm_kernel(const float* __restrict__ fac,
                                    const float* __restrict__ agg,
                                    float* __restrict__ cur)
{
    const size_t idx = (size_t)blockIdx.x * blockDim.x + threadIdx.x;
    float v = fac[idx] + agg[idx];
    float s = waveSum32(v * v);
    cur[idx] = v / fmaxf(sqrtf(s), 1e-12f);
}

// ---------------------------------------------------------------------------
// One wave per edge. lane = k within factor. Gather head(cur[row]) and
// tail(fac[col]) (512B contiguous per node), per-factor dot via wave reduce,
// softmax over FAC_K=4 factors, scatter p*tail into agg[row] via f32 atomics.
__global__ void edge_kernel(const float* __restrict__ cur,
                            const float* __restrict__ fac,
                            const int* __restrict__ rowI,
                            const int* __restrict__ colI,
                            float* __restrict__ agg,
                            int E)
{
    const int lane = threadIdx.x & 31;
    const int e = blockIdx.x * (blockDim.x >> 5) + (threadIdx.x >> 5);
    if (e >= E) return;

    const int r  = rowI[e];
    const int cn = colI[e];

    const float* hp = cur + (size_t)r  * 128 + lane;
    const float* tp = fac + (size_t)cn * 128 + lane;

    float h[FAC_K], t[FAC_K], p[FAC_K];
#pragma unroll
    for (int f = 0; f < FAC_K; ++f) { h[f] = hp[f * DIM_K]; t[f] = tp[f * DIM_K]; }

#pragma unroll
    for (int f = 0; f < FAC_K; ++f)
        p[f] = waveSum32(h[f] * t[f]);

    // softmax over factor axis (every lane has all 4 sums)
    float m = fmaxf(fmaxf(p[0], p[1]), fmaxf(p[2], p[3]));
    float s = 0.f;
#pragma unroll
    for (int f = 0; f < FAC_K; ++f) { p[f] = __expf(p[f] - m); s += p[f]; }
    const float inv = 1.f / s;

    float* ap = agg + (size_t)r * 128 + lane;
#pragma unroll
    for (int f = 0; f < FAC_K; ++f)
        atomicAdd(ap + f * DIM_K, p[f] * inv * t[f]);
}

// ---------------------------------------------------------------------------
extern "C" void kernel_launch(void* const* d_in, const int* in_sizes, int n_in,
                              void* d_out, int out_size, void* d_ws, size_t ws_size,
                              hipStream_t stream)
{
    const float* allEmb = (const float*)d_in[0];   // N x 128
    const float* W      = (const float*)d_in[1];   // 4 x 128 x 32
    const float* B      = (const float*)d_in[2];   // 4 x 1 x 32
    const int*   rowI   = (const int*)d_in[3];     // E
    const int*   colI   = (const int*)d_in[4];     // E

    const int N = in_sizes[0] / IN_DIM;            // 100000
    const int E = in_sizes[3];                     // 1600000
    const size_t total = (size_t)N * 128;          // multiple of 256

    float* cur = (float*)d_out;                    // new_fac_emb, also final output
    float* fac = (float*)d_ws;                     // normalized fac_emb
    float* agg = fac + total;                      // segment-sum accumulator

    // 1) factor GEMM + leaky_relu (WMMA f32 16x16x4)
    const int nRowTiles = N / 16;                  // 6250, exact
    gemm_fac_kernel<<<nRowTiles, 256, 0, stream>>>(allEmb, W, B, fac, nRowTiles);

    // 2) l2norm -> fac (in place) and cur
    const int nbElem = (int)(total / 256);
    norm_init_kernel<<<nbElem, 256, 0, stream>>>(fac, cur);

    // 3) ITER_K = 2 propagation iterations
    for (int it = 0; it < 2; ++it) {
        hipMemsetAsync(agg, 0, total * sizeof(float), stream);
        edge_kernel<<<(E + 7) / 8, 256, 0, stream>>>(cur, fac, rowI, colI, agg, E);
        combine_norm_kernel<<<nbElem, 256, 0, stream>>>(fac, agg, cur);
    }
}